// PScanTriton_56805237457167
// MI455X (gfx1250) — compile-verified
//
#include <hip/hip_runtime.h>

// ---------------------------------------------------------------------------
// Chunked parallel scan for y[t] = A[t] @ y[t-1] + x[t], complex 16x16 A.
//   Phase 1: per (batch,chunk) wave folds P = A_t..A_first (WMMA f32 16x16x4)
//            and q = A_t q + x_t (VALU matvec), CS steps. Double-buffered A.
//   Phase 2: per-batch wave scans the CH chunk summaries -> carry states.
//   Phase 3: per (batch,chunk) wave replays the recurrence from the carry
//            (VALU matvec only) and writes [B,L,16,2] output. Double-buffered.
// ---------------------------------------------------------------------------

typedef __attribute__((ext_vector_type(2))) float v2f;
typedef __attribute__((ext_vector_type(8))) float v8f;

constexpr int BB = 32;          // batches
constexpr int LL = 2048;        // sequence length
constexpr int CS = 32;          // chunk size (2048 waves in phases 1/3)
constexpr int CH = LL / CS;     // 64 chunks
constexpr int SUMSTRIDE = 544;  // Pr[256] Pi[256] qr[16] qi[16]

// Complex 16x16 matvec, one component per lane: lanes 0-15 -> real rows,
// lanes 16-31 -> imag rows. A rows row-major; q = [qr[16] | qi[16]].
__device__ __forceinline__ float cmatvec16(const float* Arow_r,
                                           const float* Arow_i,
                                           const float* q, int lane) {
  const int m = lane & 15;
  const bool re = lane < 16;
  const float4* ar4 = (const float4*)(Arow_r + m * 16);
  const float4* ai4 = (const float4*)(Arow_i + m * 16);
  const float4* qr4 = (const float4*)(q);
  const float4* qi4 = (const float4*)(q + 16);
  float acc = 0.f;
#pragma unroll
  for (int j = 0; j < 4; ++j) {
    const float4 a = ar4[j], b = ai4[j], u = qr4[j], v = qi4[j];
    if (re) {
      acc += a.x * u.x - b.x * v.x;
      acc += a.y * u.y - b.y * v.y;
      acc += a.z * u.z - b.z * v.z;
      acc += a.w * u.w - b.w * v.w;
    } else {
      acc += a.x * v.x + b.x * u.x;
      acc += a.y * v.y + b.y * u.y;
      acc += a.z * v.z + b.z * u.z;
      acc += a.w * v.w + b.w * u.w;
    }
  }
  return acc;
}

__device__ __forceinline__ void stage_to_lds(float* dst, const float4& r0,
                                             const float4& r1, const float4& r2,
                                             const float4& r3, int lane) {
  float4* sAr = (float4*)dst;
  float4* sAi = (float4*)(dst + 256);
  sAr[lane] = r0;
  sAr[lane + 32] = r1;
  sAi[lane] = r2;
  sAi[lane + 32] = r3;
}

// ------------------------------ Phase 1 ------------------------------------
__global__ __launch_bounds__(32) void k_chunk_summary(
    const float* __restrict__ Ar, const float* __restrict__ Ai,
    const float* __restrict__ Xr, const float* __restrict__ Xi,
    float* __restrict__ sums) {
  __shared__ __align__(16) float Abuf[2][512];  // double-buffered Ar|Ai tile
  __shared__ __align__(16) float Pbuf[512];     // Pr[16][16] | Pi[16][16]
  __shared__ __align__(16) float qbuf[32];      // qr[16] | qi[16]

  const int blk = blockIdx.x;
  const int b = blk / CH, c = blk % CH;
  const int lane = threadIdx.x;
  const int m = lane & 15, h = lane >> 4;

  // P = I, q = 0
#pragma unroll
  for (int j = 0; j < 8; ++j) {
    const int idx = lane + 32 * j;
    Pbuf[idx] = ((idx >> 4) == (idx & 15)) ? 1.f : 0.f;
    Pbuf[256 + idx] = 0.f;
  }
  qbuf[lane] = 0.f;

  const float* gAr = Ar + ((long)b * LL + (long)c * CS) * 256;
  const float* gAi = Ai + ((long)b * LL + (long)c * CS) * 256;
  const long xbase = ((long)b * LL + (long)c * CS) * 16;
  const float* Xsel = h ? Xi : Xr;

  // Prologue: stage A_0 into buffer 0.
  float4 r0, r1, r2, r3;
  float xn;
  {
    const float4* gar = (const float4*)gAr;
    const float4* gai = (const float4*)gAi;
    r0 = gar[lane];
    r1 = gar[lane + 32];
    r2 = gai[lane];
    r3 = gai[lane + 32];
    xn = Xsel[xbase + m];
  }
  stage_to_lds(Abuf[0], r0, r1, r2, r3, lane);
  __syncthreads();

  int cur = 0;
  for (int t = 0; t < CS; ++t) {
    const float x_own = xn;
    // Issue next tile's global loads early; wait/store happens after compute.
    if (t + 1 < CS) {
      const float4* gar = (const float4*)(gAr + (long)(t + 1) * 256);
      const float4* gai = (const float4*)(gAi + (long)(t + 1) * 256);
      r0 = gar[lane];
      r1 = gar[lane + 32];
      r2 = gai[lane];
      r3 = gai[lane + 32];
      xn = Xsel[xbase + (long)(t + 1) * 16 + m];
    }
    if (t + 2 < CS) {
      __builtin_prefetch(gAr + (long)(t + 2) * 256, 0, 1);
      __builtin_prefetch(gAi + (long)(t + 2) * 256, 0, 1);
    }

    const float* Ab = Abuf[cur];
    // WMMA A operands: 16x4 layout -> lane m, cols 4k+2h, 4k+2h+1 (b64).
    v2f arK[4], aiK[4], naiK[4];
#pragma unroll
    for (int k = 0; k < 4; ++k) {
      arK[k] = *(const v2f*)(Ab + m * 16 + 4 * k + 2 * h);
      aiK[k] = *(const v2f*)(Ab + 256 + m * 16 + 4 * k + 2 * h);
      naiK[k] = -aiK[k];
    }
    // WMMA B operands from P: 4x16 layout -> VGPR j holds row 4k+j+2h, col m.
    v2f prK[4], piK[4];
#pragma unroll
    for (int k = 0; k < 4; ++k) {
      v2f v;
      v.x = Pbuf[(4 * k + 0 + 2 * h) * 16 + m];
      v.y = Pbuf[(4 * k + 1 + 2 * h) * 16 + m];
      prK[k] = v;
      v.x = Pbuf[256 + (4 * k + 0 + 2 * h) * 16 + m];
      v.y = Pbuf[256 + (4 * k + 1 + 2 * h) * 16 + m];
      piK[k] = v;
    }

    // Pnew_r = Ar@Pr - Ai@Pi ; Pnew_i = Ar@Pi + Ai@Pr   (16 WMMAs, K=16)
    v8f dr = {};
    v8f di = {};
#pragma unroll
    for (int k = 0; k < 4; ++k)
      dr = __builtin_amdgcn_wmma_f32_16x16x4_f32(false, arK[k], false, prK[k],
                                                 (short)0, dr, false, false);
#pragma unroll
    for (int k = 0; k < 4; ++k)
      dr = __builtin_amdgcn_wmma_f32_16x16x4_f32(false, naiK[k], false, piK[k],
                                                 (short)0, dr, false, false);
#pragma unroll
    for (int k = 0; k < 4; ++k)
      di = __builtin_amdgcn_wmma_f32_16x16x4_f32(false, arK[k], false, piK[k],
                                                 (short)0, di, false, false);
#pragma unroll
    for (int k = 0; k < 4; ++k)
      di = __builtin_amdgcn_wmma_f32_16x16x4_f32(false, aiK[k], false, prK[k],
                                                 (short)0, di, false, false);

    // q = A_t q + x_t  (VALU matvec; WMMA would waste 14/16 columns).
    const float qn = cmatvec16(Ab, Ab + 256, qbuf, lane) + x_own;
    __syncthreads();

    // Write back P (C/D layout: reg r -> row r+8h, col m) and q; stage next A.
#pragma unroll
    for (int r = 0; r < 8; ++r) {
      Pbuf[(r + 8 * h) * 16 + m] = dr[r];
      Pbuf[256 + (r + 8 * h) * 16 + m] = di[r];
    }
    qbuf[lane] = qn;
    if (t + 1 < CS) stage_to_lds(Abuf[cur ^ 1], r0, r1, r2, r3, lane);
    __syncthreads();
    cur ^= 1;
  }

  float* out = sums + (long)blk * SUMSTRIDE;
#pragma unroll
  for (int j = 0; j < 16; ++j) out[lane + 32 * j] = Pbuf[lane + 32 * j];
  out[512 + lane] = qbuf[lane];
}

// ------------------------------ Phase 2 ------------------------------------
__global__ __launch_bounds__(32) void k_carry(const float* __restrict__ sums,
                                              float* __restrict__ carries) {
  __shared__ __align__(16) float ybuf[32];
  const int b = blockIdx.x;
  const int lane = threadIdx.x;
  ybuf[lane] = 0.f;
  __syncthreads();
  for (int c = 0; c < CH; ++c) {
    const float* s = sums + (long)(b * CH + c) * SUMSTRIDE;
    carries[(long)(b * CH + c) * 32 + lane] = ybuf[lane];  // state entering c
    float acc = cmatvec16(s, s + 256, ybuf, lane);         // P_c @ y
    acc += s[512 + lane];                                  // + q_c
    __syncthreads();
    ybuf[lane] = acc;
    __syncthreads();
  }
}

// ------------------------------ Phase 3 ------------------------------------
__global__ __launch_bounds__(32) void k_apply(
    const float* __restrict__ Ar, const float* __restrict__ Ai,
    const float* __restrict__ Xr, const float* __restrict__ Xi,
    const float* __restrict__ carries, float* __restrict__ out) {
  __shared__ __align__(16) float Abuf[2][512];
  __shared__ __align__(16) float ybuf[32];
  const int blk = blockIdx.x;
  const int b = blk / CH, c = blk % CH;
  const int lane = threadIdx.x;
  const int m = lane & 15, h = lane >> 4;

  ybuf[lane] = carries[(long)blk * 32 + lane];
  const float* gAr = Ar + ((long)b * LL + (long)c * CS) * 256;
  const float* gAi = Ai + ((long)b * LL + (long)c * CS) * 256;
  const long xbase = ((long)b * LL + (long)c * CS) * 16;
  const long obase = ((long)b * LL + (long)c * CS) * 32;
  const float* Xsel = h ? Xi : Xr;

  float4 r0, r1, r2, r3;
  float xn;
  {
    const float4* gar = (const float4*)gAr;
    const float4* gai = (const float4*)gAi;
    r0 = gar[lane];
    r1 = gar[lane + 32];
    r2 = gai[lane];
    r3 = gai[lane + 32];
    xn = Xsel[xbase + m];
  }
  stage_to_lds(Abuf[0], r0, r1, r2, r3, lane);
  __syncthreads();

  int cur = 0;
  for (int t = 0; t < CS; ++t) {
    const float x_own = xn;
    if (t + 1 < CS) {
      const float4* gar = (const float4*)(gAr + (long)(t + 1) * 256);
      const float4* gai = (const float4*)(gAi + (long)(t + 1) * 256);
      r0 = gar[lane];
      r1 = gar[lane + 32];
      r2 = gai[lane];
      r3 = gai[lane + 32];
      xn = Xsel[xbase + (long)(t + 1) * 16 + m];
    }
    if (t + 2 < CS) {
      __builtin_prefetch(gAr + (long)(t + 2) * 256, 0, 1);
      __builtin_prefetch(gAi + (long)(t + 2) * 256, 0, 1);
    }

    const float* Ab = Abuf[cur];
    const float yn = cmatvec16(Ab, Ab + 256, ybuf, lane) + x_own;
    __syncthreads();
    ybuf[lane] = yn;
    out[obase + (long)t * 32 + m * 2 + h] = yn;  // [B,L,16,2]
    if (t + 1 < CS) stage_to_lds(Abuf[cur ^ 1], r0, r1, r2, r3, lane);
    __syncthreads();
    cur ^= 1;
  }
}

// ---------------------------------------------------------------------------
extern "C" void kernel_launch(void* const* d_in, const int* in_sizes, int n_in,
                              void* d_out, int out_size, void* d_ws,
                              size_t ws_size, hipStream_t stream) {
  const float* Ar = (const float*)d_in[0];
  const float* Ai = (const float*)d_in[1];
  const float* Xr = (const float*)d_in[2];
  const float* Xi = (const float*)d_in[3];
  float* sums = (float*)d_ws;                           // 2048*544 floats
  float* carries = sums + (size_t)BB * CH * SUMSTRIDE;  // 2048*32 floats
  (void)in_sizes; (void)n_in; (void)out_size; (void)ws_size;

  k_chunk_summary<<<BB * CH, 32, 0, stream>>>(Ar, Ai, Xr, Xi, sums);
  k_carry<<<BB, 32, 0, stream>>>(sums, carries);
  k_apply<<<BB * CH, 32, 0, stream>>>(Ar, Ai, Xr, Xi, carries,
                                      (float*)d_out);
}